// FeatureEncoder_29025388987145
// MI455X (gfx1250) — compile-verified
//
#include <hip/hip_runtime.h>
#include <hip/hip_bf16.h>

// CompGCN 2-layer forward for MI455X (gfx1250, wave32, WMMA).
// Edge phase is linearized out of the per-edge GEMM: segment-sum first
// (mask-partitioned into S_O/S_I), then dense f32 WMMA GEMMs.

#define DD 64
#define N_ENT 50000
#define N_REL 237
#define N_EDGE 800000
#define BN_EPS 1e-5f

typedef __attribute__((ext_vector_type(2))) float v2f;
typedef __attribute__((ext_vector_type(8))) float v8f;

// ---------------- utility ----------------
__global__ void zero_f32(float* __restrict__ p, int n) {
    int i = blockIdx.x * blockDim.x + threadIdx.x;
    int stride = gridDim.x * blockDim.x;
    for (; i < n; i += stride) p[i] = 0.0f;
}

// ---------------- edge scatter ----------------
// 16 lanes per edge, float4 per lane. comp_h = n[src]*rel[etype]*norm,
// atomically accumulated into S_O (mask=1) or S_I (mask=0) at row dst.
__global__ void edge_scatter(const float* __restrict__ nfeat,
                             const float* __restrict__ rel,
                             const float* __restrict__ norm,
                             const int* __restrict__ src,
                             const int* __restrict__ dst,
                             const int* __restrict__ etype,
                             const int* __restrict__ omask,
                             float* __restrict__ S_O,
                             float* __restrict__ S_I) {
    long tid = (long)blockIdx.x * blockDim.x + threadIdx.x;
    int e = (int)(tid >> 4);
    if (e >= N_EDGE) return;
    int j = ((int)tid & 15) * 4;
    int s = src[e];
    int d = dst[e];
    int t = etype[e];
    float nm = norm[e];
    float* out = (omask[e] ? S_O : S_I) + (long)d * DD + j;
    float4 a = *(const float4*)(nfeat + (long)s * DD + j);
    float4 r = *(const float4*)(rel + (long)t * DD + j);
    atomicAdd(out + 0, a.x * r.x * nm);
    atomicAdd(out + 1, a.y * r.y * nm);
    atomicAdd(out + 2, a.z * r.z * nm);
    atomicAdd(out + 3, a.w * r.w * nm);
}

// ---------------- node GEMM (WMMA f32 16x16x4) ----------------
// n_pre = (S_O @ W_O.T + S_I @ W_I.T + (n ⊙ loop_rel) @ W_S.T) / 3
// Concatenated K=192. LDS weight layout Wl[n][k] so B-fragments are b64 loads.
__global__ void node_gemm(const float* __restrict__ S_O,
                          const float* __restrict__ S_I,
                          const float* __restrict__ nfeat,
                          const float* __restrict__ loop_rel,
                          const float* __restrict__ W_O,
                          const float* __restrict__ W_I,
                          const float* __restrict__ W_S,
                          float* __restrict__ n_pre) {
    __shared__ float Wl[DD][3 * DD];  // 48 KB: [n][k_cat]
    int tid = threadIdx.x;
    for (int i = tid; i < DD * DD; i += blockDim.x) {
        int n = i >> 6, k = i & 63;
        Wl[n][k] = W_O[i];
        Wl[n][64 + k] = W_I[i];
        Wl[n][128 + k] = W_S[i];
    }
    __syncthreads();

    int wave = tid >> 5;
    int lane = tid & 31;
    int tile = blockIdx.x * (blockDim.x >> 5) + wave;
    if (tile >= N_ENT / 16) return;  // wave-uniform; EXEC stays all-ones

    int m = lane & 15;
    int khalf = lane >> 4;  // 0 or 1
    long row = (long)tile * 16 + m;

    v8f acc[4] = {v8f{}, v8f{}, v8f{}, v8f{}};

    // k-range [0,64): S_O
    for (int kt = 0; kt < 16; ++kt) {
        int kk = kt * 4 + khalf * 2;
        v2f a = *(const v2f*)(S_O + row * DD + kk);
#pragma unroll
        for (int nt = 0; nt < 4; ++nt) {
            v2f b = *(const v2f*)(&Wl[nt * 16 + m][kt * 4 + khalf * 2]);
            acc[nt] = __builtin_amdgcn_wmma_f32_16x16x4_f32(
                false, a, false, b, (short)0, acc[nt], false, false);
        }
    }
    // k-range [64,128): S_I
    for (int kt = 16; kt < 32; ++kt) {
        int kk = (kt - 16) * 4 + khalf * 2;
        v2f a = *(const v2f*)(S_I + row * DD + kk);
#pragma unroll
        for (int nt = 0; nt < 4; ++nt) {
            v2f b = *(const v2f*)(&Wl[nt * 16 + m][kt * 4 + khalf * 2]);
            acc[nt] = __builtin_amdgcn_wmma_f32_16x16x4_f32(
                false, a, false, b, (short)0, acc[nt], false, false);
        }
    }
    // k-range [128,192): n ⊙ loop_rel
    for (int kt = 32; kt < 48; ++kt) {
        int kk = (kt - 32) * 4 + khalf * 2;
        v2f a;
        a.x = nfeat[row * DD + kk] * loop_rel[kk];
        a.y = nfeat[row * DD + kk + 1] * loop_rel[kk + 1];
#pragma unroll
        for (int nt = 0; nt < 4; ++nt) {
            v2f b = *(const v2f*)(&Wl[nt * 16 + m][kt * 4 + khalf * 2]);
            acc[nt] = __builtin_amdgcn_wmma_f32_16x16x4_f32(
                false, a, false, b, (short)0, acc[nt], false, false);
        }
    }

    const float inv3 = 1.0f / 3.0f;
    long mbase = (long)tile * 16 + 8 * khalf;
#pragma unroll
    for (int nt = 0; nt < 4; ++nt)
#pragma unroll
        for (int v = 0; v < 8; ++v)
            n_pre[(mbase + v) * DD + nt * 16 + m] = acc[nt][v] * inv3;
}

// ---------------- relation GEMM (WMMA f32 16x16x4) ----------------
// r_out[0:237] = (concat(rel, loop_rel) @ W_R.T)[0:237]
__global__ void rel_gemm(const float* __restrict__ rel,
                         const float* __restrict__ loop_rel,
                         const float* __restrict__ W_R,
                         float* __restrict__ r_out) {
    __shared__ float Wl[DD][DD];  // 16 KB, [n][k]
    int tid = threadIdx.x;
    for (int i = tid; i < DD * DD; i += blockDim.x) Wl[i >> 6][i & 63] = W_R[i];
    __syncthreads();

    int wave = tid >> 5;
    int lane = tid & 31;
    int tile = blockIdx.x * (blockDim.x >> 5) + wave;
    if (tile >= 15) return;  // ceil(238/16)

    int m = lane & 15;
    int khalf = lane >> 4;
    int row = tile * 16 + m;
    // rows >= 237 read loop_rel (row 237 genuinely is it; 238/239 are masked on store)
    const float* xrow = (row < N_REL) ? (rel + (long)row * DD) : loop_rel;

    v8f acc[4] = {v8f{}, v8f{}, v8f{}, v8f{}};
    for (int kt = 0; kt < 16; ++kt) {
        v2f a = *(const v2f*)(xrow + kt * 4 + khalf * 2);
#pragma unroll
        for (int nt = 0; nt < 4; ++nt) {
            v2f b = *(const v2f*)(&Wl[nt * 16 + m][kt * 4 + khalf * 2]);
            acc[nt] = __builtin_amdgcn_wmma_f32_16x16x4_f32(
                false, a, false, b, (short)0, acc[nt], false, false);
        }
    }
#pragma unroll
    for (int nt = 0; nt < 4; ++nt)
#pragma unroll
        for (int v = 0; v < 8; ++v) {
            int ro = tile * 16 + 8 * khalf + v;
            if (ro < N_REL) r_out[(long)ro * DD + nt * 16 + m] = acc[nt][v];
        }
}

// ---------------- batchnorm ----------------
__global__ void bn_stats(const float* __restrict__ x,
                         float* __restrict__ gsum, float* __restrict__ gsq) {
    __shared__ float ssum[256], ssq[256];
    int tid = threadIdx.x;
    int c = tid & 63;
    int rg = tid >> 6;  // 0..3
    float s = 0.0f, s2 = 0.0f;
    for (int r = blockIdx.x * 4 + rg; r < N_ENT; r += gridDim.x * 4) {
        float v = x[(long)r * DD + c];
        s += v;
        s2 += v * v;
    }
    ssum[tid] = s;
    ssq[tid] = s2;
    __syncthreads();
    if (tid < 64) {
        s = ssum[tid] + ssum[tid + 64] + ssum[tid + 128] + ssum[tid + 192];
        s2 = ssq[tid] + ssq[tid + 64] + ssq[tid + 128] + ssq[tid + 192];
        atomicAdd(&gsum[c], s);
        atomicAdd(&gsq[c], s2);
    }
}

__global__ void bn_params(const float* __restrict__ gsum, const float* __restrict__ gsq,
                          const float* __restrict__ gamma, const float* __restrict__ beta,
                          float* __restrict__ scale, float* __restrict__ shift) {
    int c = threadIdx.x;
    if (c < DD) {
        float mean = gsum[c] * (1.0f / N_ENT);
        float var = gsq[c] * (1.0f / N_ENT) - mean * mean;
        float sc = gamma[c] * rsqrtf(var + BN_EPS);
        scale[c] = sc;
        shift[c] = beta[c] - mean * sc;
    }
}

__global__ void bn_apply_tanh(const float* __restrict__ x,
                              const float* __restrict__ scale,
                              const float* __restrict__ shift,
                              float* __restrict__ y) {
    int i = blockIdx.x * blockDim.x + threadIdx.x;
    int stride = gridDim.x * blockDim.x;
    for (; i < N_ENT * DD; i += stride) {
        int c = i & 63;
        y[i] = tanhf(x[i] * scale[c] + shift[c]);
    }
}

// ---------------- driver ----------------
extern "C" void kernel_launch(void* const* d_in, const int* in_sizes, int n_in,
                              void* d_out, int out_size, void* d_ws, size_t ws_size,
                              hipStream_t stream) {
    const float* n_embds = (const float*)d_in[0];
    const float* rel_embds = (const float*)d_in[1];
    const float* norm = (const float*)d_in[2];
    const int* src = (const int*)d_in[3];
    const int* dst = (const int*)d_in[4];
    const int* etype = (const int*)d_in[5];
    const int* omask = (const int*)d_in[6];
    const float* W[2][4];
    const float* lr[2];
    const float* gm[2];
    const float* bt[2];
    for (int l = 0; l < 2; ++l) {
        int b = 7 + l * 7;
        W[l][0] = (const float*)d_in[b + 0];  // W_O
        W[l][1] = (const float*)d_in[b + 1];  // W_I
        W[l][2] = (const float*)d_in[b + 2];  // W_S
        W[l][3] = (const float*)d_in[b + 3];  // W_R
        lr[l] = (const float*)d_in[b + 4];
        gm[l] = (const float*)d_in[b + 5];
        bt[l] = (const float*)d_in[b + 6];
    }

    float* ws = (float*)d_ws;
    const long NF = (long)N_ENT * DD;       // 3,200,000
    float* S_O = ws;                        // [N_ENT, 64]
    float* S_I = ws + NF;                   // [N_ENT, 64]
    float* n_pre = ws + 2 * NF;             // [N_ENT, 64]
    float* n1 = ws + 3 * NF;                // layer-1 node output
    float* r1 = ws + 4 * NF;                // [237, 64]
    float* gsum = r1 + (long)N_REL * DD;
    float* gsq = gsum + 64;
    float* bscale = gsq + 64;
    float* bshift = bscale + 64;

    float* out_n = (float*)d_out;                 // [N_ENT, 64]
    float* out_r = (float*)d_out + NF;            // [237, 64]

    const int ntiles = N_ENT / 16;                // 3125
    const int gemm_blocks = (ntiles + 7) / 8;     // 391
    const int edge_blocks = (N_EDGE * 16 + 255) / 256;

    const float* nf_in = n_embds;
    const float* rf_in = rel_embds;
    for (int l = 0; l < 2; ++l) {
        float* n_out_l = (l == 0) ? n1 : out_n;
        float* r_out_l = (l == 0) ? r1 : out_r;

        zero_f32<<<4096, 256, 0, stream>>>(S_O, (int)(2 * NF));  // S_O + S_I contiguous
        zero_f32<<<1, 256, 0, stream>>>(gsum, 128);

        edge_scatter<<<edge_blocks, 256, 0, stream>>>(nf_in, rf_in, norm, src, dst,
                                                      etype, omask, S_O, S_I);
        node_gemm<<<gemm_blocks, 256, 0, stream>>>(S_O, S_I, nf_in, lr[l],
                                                   W[l][0], W[l][1], W[l][2], n_pre);
        bn_stats<<<512, 256, 0, stream>>>(n_pre, gsum, gsq);
        bn_params<<<1, 64, 0, stream>>>(gsum, gsq, gm[l], bt[l], bscale, bshift);
        bn_apply_tanh<<<4096, 256, 0, stream>>>(n_pre, bscale, bshift, n_out_l);
        rel_gemm<<<2, 256, 0, stream>>>(rf_in, lr[l], W[l][3], r_out_l);

        nf_in = n_out_l;
        rf_in = r_out_l;
    }
}